// SBEceLoss_76209899700731
// MI455X (gfx1250) — compile-verified
//
#include <hip/hip_runtime.h>
#include <math.h>

// ---------------------------------------------------------------------------
// SBEceLoss for MI455X (gfx1250, wave32)
//   K1: wave-per-row streaming log-softmax max/argmax/lse (HBM-bound, ~9us floor)
//       - float4 (global_load_b128) fast path for C%4==0 && C<=128
//       - global_prefetch_b8 of next row via __builtin_prefetch
//   K2: soft-binning reduction as [3 x N] x [N x 15] matmul via V_WMMA_F32_16X16X4_F32
//       - fully branchless inner loop (unconditional ALU, 0/1-mask multiplies)
//   K3: scalar ECE epilogue
// ---------------------------------------------------------------------------

typedef __attribute__((ext_vector_type(2))) float v2f;
typedef __attribute__((ext_vector_type(8))) float v8f;

#define N_BINS 15
#define INV_TEMP 100.0f
#define EPS 1e-5f
#define NEG_FLT_MAX (-3.402823466e38f)

// ---------------------------------------------------------------------------
// Kernel 1a: fast path, one float4 per lane covers the row (C%4==0, C<=128).
// Two register passes: (max, argmax) then sum exp(x - max). No divergence.
// ---------------------------------------------------------------------------
__global__ void sbece_rowstat_f4(const float* __restrict__ logits,
                                 const long long* __restrict__ labels,
                                 float* __restrict__ conf_out,
                                 float* __restrict__ acc_out,
                                 int n_rows, int n_cols) {
  const int lane = threadIdx.x & 31;
  const int wave = threadIdx.x >> 5;
  const int waves_per_block = blockDim.x >> 5;
  const int wave_id = blockIdx.x * waves_per_block + wave;
  const int n_waves = gridDim.x * waves_per_block;

  const int nvec = n_cols >> 2;           // float4s per row
  const bool act = lane < nvec;
  const int sl = act ? lane : 0;          // safe vector index (always in-bounds)

  for (int r = wave_id; r < n_rows; r += n_waves) {
    const float4* rowv = (const float4*)(logits + (size_t)r * (size_t)n_cols);

    // prefetch the next row this wave will stream (global_prefetch_b8)
    const int rn = r + n_waves;
    if (rn < n_rows) {
      const float4* nx = (const float4*)(logits + (size_t)rn * (size_t)n_cols);
      __builtin_prefetch(nx + sl, 0, 1);
    }

    float4 x = rowv[sl];                  // one global_load_b128 per lane
    float x0 = x.x, x1 = x.y, x2 = x.z, x3 = x.w;
    if (!act) { x0 = x1 = x2 = x3 = NEG_FLT_MAX; }

    // pass 1: lane-local max/argmax (first occurrence = lowest index)
    const int ib = sl << 2;
    float m = x0; int mi = ib;
    if (x1 > m) { m = x1; mi = ib + 1; }
    if (x2 > m) { m = x2; mi = ib + 2; }
    if (x3 > m) { m = x3; mi = ib + 3; }
    if (!act) mi = 0x7fffffff;

    // cross-lane (max, min-index-on-tie) butterfly — fixed order, deterministic
    #pragma unroll
    for (int off = 16; off > 0; off >>= 1) {
      float om = __shfl_xor(m, off, 32);
      int   oi = __shfl_xor(mi, off, 32);
      if (om > m || (om == m && oi < mi)) { mi = oi; }
      m = fmaxf(m, om);
    }

    // pass 2: shifted exp-sum (registers only; 1 v_exp_f32 per element)
    float s = act ? (__expf(x0 - m) + __expf(x1 - m) +
                     __expf(x2 - m) + __expf(x3 - m)) : 0.0f;
    #pragma unroll
    for (int off = 16; off > 0; off >>= 1) s += __shfl_xor(s, off, 32);

    if (lane == 0) {
      float pred = -__logf(s);            // max(log_softmax) = -log(sum exp(x-max))
      float lab = (float)labels[r];
      conf_out[r] = (float)mi;
      acc_out[r]  = (pred == lab) ? 1.0f : 0.0f;  // faithful to reference
    }
  }
}

// ---------------------------------------------------------------------------
// Kernel 1b: generic fallback (any C), single-pass online softmax per lane.
// ---------------------------------------------------------------------------
__global__ void sbece_rowstat_gen(const float* __restrict__ logits,
                                  const long long* __restrict__ labels,
                                  float* __restrict__ conf_out,
                                  float* __restrict__ acc_out,
                                  int n_rows, int n_cols) {
  const int lane = threadIdx.x & 31;
  const int wave = threadIdx.x >> 5;
  const int waves_per_block = blockDim.x >> 5;
  const int wave_id = blockIdx.x * waves_per_block + wave;
  const int n_waves = gridDim.x * waves_per_block;

  for (int r = wave_id; r < n_rows; r += n_waves) {
    const float* row = logits + (size_t)r * (size_t)n_cols;
    float m = NEG_FLT_MAX, s = 0.0f;
    int mi = 0x7fffffff;
    for (int j = lane; j < n_cols; j += 32) {
      float x = row[j];
      if (x > m) { s = s * __expf(m - x) + 1.0f; m = x; mi = j; }
      else       { s += __expf(x - m); }
    }
    #pragma unroll
    for (int off = 16; off > 0; off >>= 1) {
      float om = __shfl_xor(m, off, 32);
      float os = __shfl_xor(s, off, 32);
      int   oi = __shfl_xor(mi, off, 32);
      float nm = fmaxf(m, om);
      s = s * __expf(m - nm) + os * __expf(om - nm);
      if (om > m || (om == m && oi < mi)) mi = oi;
      m = nm;
    }
    if (lane == 0) {
      float pred = -__logf(s);
      conf_out[r] = (float)mi;
      acc_out[r]  = (pred == (float)labels[r]) ? 1.0f : 0.0f;
    }
  }
}

// unnormalized, max-stabilized soft-bin weight: exp(d_bin - d_nearest_anchor)
__device__ __forceinline__ float sbece_binw(float cf, float abin) {
  float bn = fminf(fmaxf(rintf(cf * 15.0f - 0.5f), 0.0f), 14.0f);
  float an = (bn + 0.5f) * (1.0f / 15.0f);
  float db = cf - abin, dn = cf - an;
  return __expf((dn * dn - db * db) * INV_TEMP);
}

// ---------------------------------------------------------------------------
// Kernel 2: bin accumulation via WMMA f32 16x16x4.
//   A (16x4): col k = [1, conf_k, acc_k, 0...]^T     (M = lane%16, K = 2*(lane/16)+v)
//   B (4x16): row k = coeffs_k[bin], bin 15 padded 0 (N = lane%16, K = 2*(lane/16)+v)
//   D (16x16) accumulates: row0 = sum(coeff), row1 = sum(conf*coeff), row2 = sum(acc*coeff)
// Branchless: clamped loads; exp chains computed unconditionally; validity folded
// in as 0/1 multiplies so nothing gets sunk under a divergent branch.
// ---------------------------------------------------------------------------
__global__ void sbece_binacc(const float* __restrict__ conf,
                             const float* __restrict__ acc,
                             float* __restrict__ partials,
                             int n_rows) {
  __shared__ float lds[8 * 48];

  const int lane = threadIdx.x & 31;
  const int wave = threadIdx.x >> 5;
  const int waves_per_block = blockDim.x >> 5;
  const int wave_id = blockIdx.x * waves_per_block + wave;
  const int n_waves = gridDim.x * waves_per_block;

  const int grp = lane >> 4;        // 0: K rows {0,1}; 1: K rows {2,3}
  const int mn  = lane & 15;        // M for A, bin index for B (15 = zero pad)
  const int k0  = 2 * grp;
  const int k1  = 2 * grp + 1;
  const float abin = ((float)mn + 0.5f) * (1.0f / 15.0f);
  const float binmask = (mn < N_BINS) ? 1.0f : 0.0f;  // lane 15/31 = pad column
  const int last = n_rows - 1;

  v8f c = {};

  for (int base = wave_id * 4; base < n_rows; base += n_waves * 4) {
    const int r0 = base + k0;
    const int r1 = base + k1;
    const float g0 = (r0 < n_rows) ? 1.0f : 0.0f;   // row validity as 0/1 factor
    const float g1 = (r1 < n_rows) ? 1.0f : 0.0f;
    const int i0 = (r0 < n_rows) ? r0 : last;       // clamped -> loads always safe
    const int i1 = (r1 < n_rows) ? r1 : last;

    float c0 = conf[i0], c1 = conf[i1];
    float a0 = acc[i0],  a1 = acc[i1];

    // unconditional ALU, masked by multiply (keeps the loop straight-line)
    float e0 = sbece_binw(c0, abin) * binmask * g0;
    float e1 = sbece_binw(c1, abin) * binmask * g1;

    // softmax denominator: sum over the 16-lane half (bins 0..14 + zero pad)
    float s0 = e0, s1 = e1;
    #pragma unroll
    for (int off = 8; off > 0; off >>= 1) {
      s0 += __shfl_xor(s0, off, 16);
      s1 += __shfl_xor(s1, off, 16);
    }
    float n0 = (s0 > 0.0f) ? (1.0f / s0) : 0.0f;   // invalid rows -> zero B column
    float n1 = (s1 > 0.0f) ? (1.0f / s1) : 0.0f;

    v2f A, B;
    A[0] = ((mn == 0) ? 1.0f : (mn == 1) ? c0 : (mn == 2) ? a0 : 0.0f) * g0;
    A[1] = ((mn == 0) ? 1.0f : (mn == 1) ? c1 : (mn == 2) ? a1 : 0.0f) * g1;
    B[0] = e0 * n0;
    B[1] = e1 * n1;

    c = __builtin_amdgcn_wmma_f32_16x16x4_f32(false, A, false, B,
                                              (short)0, c, false, false);
  }

  // D rows 0..2 live in c[0..2] on lanes 0..15 (N = lane)
  if (lane < 16) {
    lds[wave * 48 +      lane] = c[0];   // sum coeff        per bin
    lds[wave * 48 + 16 + lane] = c[1];   // sum conf*coeff   per bin
    lds[wave * 48 + 32 + lane] = c[2];   // sum acc*coeff    per bin
  }
  __syncthreads();

  // deterministic fixed-order cross-wave sum, one slot per thread
  if (threadIdx.x < 48) {
    float t = 0.0f;
    for (int w = 0; w < waves_per_block; ++w) t += lds[w * 48 + threadIdx.x];
    partials[blockIdx.x * 48 + threadIdx.x] = t;
  }
}

// ---------------------------------------------------------------------------
// Kernel 3: deterministic finalize -> scalar ECE
// ---------------------------------------------------------------------------
__global__ void sbece_finalize(const float* __restrict__ partials,
                               int nblocks, float* __restrict__ out) {
  __shared__ float tot[48];
  const int t = threadIdx.x;
  if (t < 48) {
    float s = 0.0f;
    for (int b = 0; b < nblocks; ++b) s += partials[b * 48 + t];
    tot[t] = s;
  }
  __syncthreads();
  if (t == 0) {
    float sumabs = 0.0f;
    for (int b = 0; b < N_BINS; ++b) sumabs += fabsf(tot[b]);
    float l1 = fmaxf(sumabs, EPS);
    float acc2 = 0.0f;
    for (int b = 0; b < N_BINS; ++b) {
      float S  = tot[b];
      float dn = fmaxf(S, EPS);
      float bc = tot[16 + b] / dn;
      float ba = tot[32 + b] / dn;
      float d  = bc - ba;
      acc2 += d * d * (S / l1);
    }
    out[0] = sqrtf(acc2);
  }
}

// ---------------------------------------------------------------------------
extern "C" void kernel_launch(void* const* d_in, const int* in_sizes, int n_in,
                              void* d_out, int out_size, void* d_ws, size_t ws_size,
                              hipStream_t stream) {
  const float* logits = (const float*)d_in[0];
  const long long* labels = (const long long*)d_in[1];
  const int N = in_sizes[1];
  const int C = in_sizes[0] / N;

  float* conf = (float*)d_ws;          // N floats
  float* acc  = conf + N;              // N floats
  float* part = acc + N;               // NB2*48 floats

  const int NB2 = 512;

  if ((C & 3) == 0 && C <= 128) {
    sbece_rowstat_f4<<<2048, 256, 0, stream>>>(logits, labels, conf, acc, N, C);
  } else {
    sbece_rowstat_gen<<<2048, 256, 0, stream>>>(logits, labels, conf, acc, N, C);
  }
  sbece_binacc<<<NB2, 256, 0, stream>>>(conf, acc, part, N);
  sbece_finalize<<<1, 64, 0, stream>>>(part, NB2, (float*)d_out);
}